// NewsEncoder_36378372997626
// MI455X (gfx1250) — compile-verified
//
#include <hip/hip_runtime.h>
#include <hip/hip_bf16.h>

typedef __attribute__((ext_vector_type(2))) float v2f;
typedef __attribute__((ext_vector_type(8))) float v8f;

#define CATV   1000
#define TYPEV  10
#define EMB    64
#define PROJ   16
#define MAXSUB 8
#define SENT   3
#define W1OUT  32
#define HIN    37
#define KPAD   40
#define XSTR   41   // padded LDS row stride (floats), odd -> bank friendly

// ---------------------------------------------------------------------------
// Kernel 1: fold the 64-wide embedding dim away once.
//   P[1000][16] = cat_emb @ cat_W          (no bias; 2*cat_b added later)
//   T[10][16]   = type_emb @ type_W + type_b
// ---------------------------------------------------------------------------
__global__ void precompute_kernel(const float* __restrict__ cat_emb,
                                  const float* __restrict__ cat_W,
                                  const float* __restrict__ type_emb,
                                  const float* __restrict__ type_W,
                                  const float* __restrict__ type_b,
                                  float* __restrict__ P,
                                  float* __restrict__ T) {
    int idx = blockIdx.x * blockDim.x + threadIdx.x;
    if (idx < CATV * PROJ) {
        int row = idx >> 4, col = idx & 15;
        float s = 0.f;
        #pragma unroll 8
        for (int e = 0; e < EMB; ++e)
            s += cat_emb[row * EMB + e] * cat_W[e * PROJ + col];
        P[idx] = s;
    } else if (idx < CATV * PROJ + TYPEV * PROJ) {
        int i = idx - CATV * PROJ;
        int row = i >> 4, col = i & 15;
        float s = type_b[col];
        #pragma unroll 8
        for (int e = 0; e < EMB; ++e)
            s += type_emb[row * EMB + e] * type_W[e * PROJ + col];
        T[i] = s;
    }
}

// ---------------------------------------------------------------------------
// Shared helper: ct[16] = P[cat] + (sum_masked P[sub])/len + 2*cat_b
// ---------------------------------------------------------------------------
__device__ __forceinline__ void cat_feature(int r,
                                            const int* __restrict__ cat,
                                            const int* __restrict__ sub,
                                            const int* __restrict__ sub_len,
                                            const float* __restrict__ P,
                                            const float* __restrict__ cat_b,
                                            float ct[PROJ]) {
    int c   = cat[r];
    int len = sub_len[r];
    const float* pb = P + c * PROJ;
    float acc[PROJ];
    #pragma unroll
    for (int j = 0; j < PROJ; ++j) acc[j] = 0.f;
    #pragma unroll
    for (int s = 0; s < MAXSUB; ++s) {
        int si = sub[r * MAXSUB + s];          // always a valid index
        const float* ps = P + si * PROJ;
        float m = (s < len) ? 1.f : 0.f;
        #pragma unroll
        for (int j = 0; j < PROJ; ++j) acc[j] += m * ps[j];
    }
    float inv = 1.f / (float)((len < 1) ? 1 : len);
    #pragma unroll
    for (int j = 0; j < PROJ; ++j)
        ct[j] = pb[j] + acc[j] * inv + 2.f * cat_b[j];
}

// ---------------------------------------------------------------------------
// Kernel 2: target branch -> x_t [Nt, 35]
// ---------------------------------------------------------------------------
__global__ void target_kernel(const int* __restrict__ cat_t,
                              const int* __restrict__ sub_t,
                              const int* __restrict__ sub_len_t,
                              const float* __restrict__ sent_t,
                              const int* __restrict__ type_t,
                              const float* __restrict__ P,
                              const float* __restrict__ T,
                              const float* __restrict__ cat_b,
                              float* __restrict__ out, int Nt) {
    int r = blockIdx.x * blockDim.x + threadIdx.x;
    if (r >= Nt) return;
    float ct[PROJ];
    cat_feature(r, cat_t, sub_t, sub_len_t, P, cat_b, ct);
    float* o = out + (size_t)r * 35;
    #pragma unroll
    for (int j = 0; j < PROJ; ++j) o[j] = ct[j];
    o[16] = sent_t[r * 3 + 0];
    o[17] = sent_t[r * 3 + 1];
    o[18] = sent_t[r * 3 + 2];
    const float* t = T + type_t[r] * PROJ;
    #pragma unroll
    for (int j = 0; j < PROJ; ++j) o[19 + j] = t[j];
}

// ---------------------------------------------------------------------------
// Kernel 3: history branch. One wave per 32 rows (two 16-row M tiles).
// Features assembled in per-wave LDS, projected with v_wmma_f32_16x16x4_f32.
// ---------------------------------------------------------------------------
__global__ __launch_bounds__(256) void hist_kernel(
        const int*   __restrict__ cat_h,
        const int*   __restrict__ sub_h,
        const int*   __restrict__ sub_len_h,
        const float* __restrict__ sent_h,
        const int*   __restrict__ type_h,
        const float* __restrict__ rt,
        const float* __restrict__ sc,
        const float* __restrict__ P,
        const float* __restrict__ T,
        const float* __restrict__ cat_b,
        const float* __restrict__ w1_W,
        const float* __restrict__ w1_b,
        float* __restrict__ out, int nChunks) {
    __shared__ float xs[8][32 * XSTR];

    const int lane = threadIdx.x & 31;
    const int wave = threadIdx.x >> 5;
    const int hi   = lane >> 4;    // lane half
    const int lm   = lane & 15;

    // --- loop-invariant B fragments for both N tiles, all 10 K-steps -------
    // B layout (4x16, KxN): VGPR0 = row K=(0|2 by half), VGPR1 = K=(1|3).
    v2f bf[10][2];
    #pragma unroll
    for (int s = 0; s < 10; ++s) {
        int k = 4 * s + 2 * hi;
        #pragma unroll
        for (int nt = 0; nt < 2; ++nt) {
            int n = nt * 16 + lm;
            v2f b;
            b.x = (k     < HIN) ? w1_W[k * W1OUT + n]       : 0.f;
            b.y = (k + 1 < HIN) ? w1_W[(k + 1) * W1OUT + n] : 0.f;
            bf[s][nt] = b;
        }
    }
    const float wb0 = w1_b[lm];
    const float wb1 = w1_b[16 + lm];

    float* xrowW = &xs[wave][lane * XSTR];

    for (int chunk = blockIdx.x * 8 + wave; chunk < nChunks;
         chunk += gridDim.x * 8) {
        int r = chunk * 32 + lane;               // nChunks*32 == Nh

        // ---- assemble padded 40-float feature row in LDS ------------------
        float ct[PROJ];
        cat_feature(r, cat_h, sub_h, sub_len_h, P, cat_b, ct);
        #pragma unroll
        for (int j = 0; j < PROJ; ++j) xrowW[j] = ct[j];
        xrowW[16] = sent_h[r * 3 + 0];
        xrowW[17] = sent_h[r * 3 + 1];
        xrowW[18] = sent_h[r * 3 + 2];
        const float* t = T + type_h[r] * PROJ;
        #pragma unroll
        for (int j = 0; j < PROJ; ++j) xrowW[19 + j] = t[j];
        xrowW[35] = rt[r];
        xrowW[36] = sc[r];
        xrowW[37] = 0.f; xrowW[38] = 0.f; xrowW[39] = 0.f;

        // per-wave LDS scratch: only need this wave's DS ops complete
        asm volatile("s_wait_dscnt 0x0" ::: "memory");

        // ---- two 16-row M tiles x two 16-col N tiles ----------------------
        #pragma unroll
        for (int mt = 0; mt < 2; ++mt) {
            const float* xrow = &xs[wave][(mt * 16 + lm) * XSTR];
            v8f c0 = {};
            v8f c1 = {};
            #pragma unroll
            for (int s = 0; s < 10; ++s) {
                int k = 4 * s + 2 * hi;
                // A layout (16x4): VGPR0 = K=(0|2 by half), VGPR1 = K=(1|3)
                v2f a;
                a.x = xrow[k];
                a.y = xrow[k + 1];
                c0 = __builtin_amdgcn_wmma_f32_16x16x4_f32(
                        false, a, false, bf[s][0], (short)0, c0, false, false);
                c1 = __builtin_amdgcn_wmma_f32_16x16x4_f32(
                        false, a, false, bf[s][1], (short)0, c1, false, false);
            }
            // D layout: VGPR v -> row (v + 8*hi), col lm
            int baseRow = chunk * 32 + mt * 16 + 8 * hi;
            #pragma unroll
            for (int v = 0; v < 8; ++v) {
                size_t m = (size_t)(baseRow + v) * W1OUT;
                out[m + lm]      = c0[v] + wb0;
                out[m + 16 + lm] = c1[v] + wb1;
            }
        }
        asm volatile("s_wait_dscnt 0x0" ::: "memory");  // before next overwrite
    }
}

// ---------------------------------------------------------------------------
extern "C" void kernel_launch(void* const* d_in, const int* in_sizes, int n_in,
                              void* d_out, int out_size, void* d_ws, size_t ws_size,
                              hipStream_t stream) {
    const int*   cat_t     = (const int*)  d_in[0];
    const int*   sub_t     = (const int*)  d_in[1];
    const int*   sub_len_t = (const int*)  d_in[2];
    const float* sent_t    = (const float*)d_in[3];
    const int*   type_t    = (const int*)  d_in[4];
    const int*   cat_h     = (const int*)  d_in[5];
    const int*   sub_h     = (const int*)  d_in[6];
    const int*   sub_len_h = (const int*)  d_in[7];
    const float* sent_h    = (const float*)d_in[8];
    const int*   type_h    = (const int*)  d_in[9];
    const float* rt        = (const float*)d_in[10];
    const float* sc        = (const float*)d_in[11];
    const float* cat_emb   = (const float*)d_in[12];
    const float* cat_W     = (const float*)d_in[13];
    const float* cat_b     = (const float*)d_in[14];
    const float* type_emb  = (const float*)d_in[15];
    const float* type_W    = (const float*)d_in[16];
    const float* type_b    = (const float*)d_in[17];
    const float* w1_W      = (const float*)d_in[18];
    const float* w1_b      = (const float*)d_in[19];

    const int Nt = in_sizes[0];
    const int Nh = in_sizes[5];

    float* P = (float*)d_ws;                 // 1000*16 floats
    float* T = P + CATV * PROJ;              // 10*16 floats

    float* out_t = (float*)d_out;            // [Nt, 35]
    float* out_h = out_t + (size_t)Nt * 35;  // [Nh, 32]

    // 1) fold embedding dim
    {
        int total = CATV * PROJ + TYPEV * PROJ;
        int blocks = (total + 255) / 256;
        precompute_kernel<<<blocks, 256, 0, stream>>>(cat_emb, cat_W, type_emb,
                                                      type_W, type_b, P, T);
    }
    // 2) target branch
    {
        int blocks = (Nt + 255) / 256;
        target_kernel<<<blocks, 256, 0, stream>>>(cat_t, sub_t, sub_len_t,
                                                  sent_t, type_t, P, T, cat_b,
                                                  out_t, Nt);
    }
    // 3) history branch (WMMA); Nh is a multiple of 32
    {
        int nChunks = Nh / 32;
        int blocks  = (nChunks + 7) / 8;
        if (blocks > 1024) blocks = 1024;
        hist_kernel<<<blocks, 256, 0, stream>>>(cat_h, sub_h, sub_len_h, sent_h,
                                                type_h, rt, sc, P, T, cat_b,
                                                w1_W, w1_b, out_h, nChunks);
    }
}